// MyLossFunction_15839839387745
// MI455X (gfx1250) — compile-verified
//
#include <hip/hip_runtime.h>
#include <math.h>
#include <stdint.h>

#define NT 16
#define NR 8
#define NS 32
#define BATCH 32768
#define WAVES 4
#define NBLOCKS (BATCH / WAVES)
#define ROW_LEN 900   // 4 + 512 + 256 + 128

typedef float v2f __attribute__((ext_vector_type(2)));
typedef float v8f __attribute__((ext_vector_type(8)));

struct alignas(16) WS {
  float H[NS * NT];     // offset 0    (16B aligned)
  float Gr[16 * NS];    // offset 2048 (16B aligned); rows 8..15 zero pad
  float Hb[NR * NT];    // offset 4096 (16B aligned)
  float cs[NS], sn[NS];
  float Mr[NR * NT], Mi[NR * NT];
  float A1r[NT * NT], A1i[NT * NT];
  float L[NT * NT];
  float Zr[NT * NT], Zi[NT * NT];
  float Cr[NT * NT], Ci[NT * NT];
  float vr[NT], vi[NT];
  float br[NT], bi[NT];
  float red[32];
};

__device__ __forceinline__ v8f wmma_f32(v2f a, v2f b, v8f c) {
  // D = A(16x4) * B(4x16) + C ; f32 WMMA, wave32
  return __builtin_amdgcn_wmma_f32_16x16x4_f32(false, a, false, b, (short)0, c,
                                               false, false);
}

// async copy of one 16B chunk: global -> LDS (ASYNCcnt-tracked)
__device__ __forceinline__ void async_b128(uint32_t lds_off, const void* gsrc) {
  asm volatile("global_load_async_to_lds_b128 %0, %1, off"
               :
               : "v"(lds_off), "v"(gsrc)
               : "memory");
}

__device__ __forceinline__ uint32_t lds_off_of(const void* p) {
  // LDS aperture base has zero low 32 bits: truncation yields the DS offset.
  return (uint32_t)(uintptr_t)p;
}

__global__ __launch_bounds__(32 * WAVES) void beam_loss_kernel(
    const float* __restrict__ phase, const float* __restrict__ target,
    float* __restrict__ partial) {
  __shared__ WS smem[WAVES];
  __shared__ float blockAcc[WAVES];

  const int tid = threadIdx.x;
  const int wave = tid >> 5;
  const int lane = tid & 31;
  const int m = lane & 15;   // M (or N) index for WMMA operands
  const int h = lane >> 4;   // lane half
  const int b = blockIdx.x * WAVES + wave;
  WS& s = smem[wave];

  const float* trow = target + (size_t)b * ROW_LEN;
  const float alpha = trow[0];
  const float beta = trow[1];
  const float SNR = trow[2];
  const float Rs = trow[3];
  const float a2 = alpha * alpha;
  const float b2 = beta * beta;

  // ---------------- stage item into LDS (async-to-LDS path) ----------------
  {
    const uint32_t ldsH = lds_off_of(s.H);
    const char* gH = (const char*)(trow + 4);        // 2048 B, 16B aligned
#pragma unroll
    for (int c = 0; c < 4; ++c) {
      const int idx = c * 32 + lane;                  // 0..127
      async_b128(ldsH + idx * 16, gH + idx * 16);
    }
    const uint32_t ldsG = lds_off_of(s.Gr);
    const char* gG = (const char*)(trow + 516);      // 1024 B
#pragma unroll
    for (int c = 0; c < 2; ++c) {
      const int idx = c * 32 + lane;                  // 0..63
      async_b128(ldsG + idx * 16, gG + idx * 16);
    }
    const uint32_t ldsHb = lds_off_of(s.Hb);
    const char* gHb = (const char*)(trow + 772);     // 512 B
    async_b128(ldsHb + lane * 16, gHb + lane * 16);
  }
  for (int i = lane; i < NR * NS; i += 32) s.Gr[256 + i] = 0.0f;  // pad rows
  {
    float ph = phase[(size_t)b * NS + lane] * 6.28318530717958647692f;
    s.cs[lane] = cosf(ph);
    s.sn[lane] = sinf(ph);
  }
  asm volatile("s_wait_asynccnt 0" ::: "memory");
  __syncthreads();

  // ---------------- M = Hb + Gr(16x32 padded) * PhiH(32x16)  (WMMA) --------
  {
    v8f accR = {};
    v8f accI = {};
#pragma unroll
    for (int kk = 0; kk < 8; ++kk) {
      const int k0 = kk * 4 + 2 * h;
      const int k1 = k0 + 1;
      v2f a;
      a.x = s.Gr[m * NS + k0];
      a.y = s.Gr[m * NS + k1];
      v2f bR, bI;
      const float h0 = s.H[k0 * NT + m];
      const float h1 = s.H[k1 * NT + m];
      bR.x = s.cs[k0] * h0;
      bR.y = s.cs[k1] * h1;
      bI.x = s.sn[k0] * h0;
      bI.y = s.sn[k1] * h1;
      accR = wmma_f32(a, bR, accR);
      accI = wmma_f32(a, bI, accI);
    }
    if (h == 0) {  // D rows 0..7 live in lane-half 0
#pragma unroll 8
      for (int g = 0; g < 8; ++g) {
        s.Mr[g * NT + m] = accR[g] + s.Hb[g * NT + m];
        s.Mi[g * NT + m] = accI[g];
      }
    }
  }
  __syncthreads();

  // ---------------- A1 = M^H M + t I  (WMMA, K=8) --------------------------
  {
    v8f pR = {};
    v8f pI = {};
#pragma unroll
    for (int kk = 0; kk < 2; ++kk) {
      const int k0 = kk * 4 + 2 * h;
      const int k1 = k0 + 1;
      v2f ar, ai, nai;
      ar.x = s.Mr[k0 * NT + m];
      ar.y = s.Mr[k1 * NT + m];
      ai.x = s.Mi[k0 * NT + m];
      ai.y = s.Mi[k1 * NT + m];
      nai.x = -ai.x;
      nai.y = -ai.y;
      pR = wmma_f32(ar, ar, pR);   // Mr^T Mr
      pR = wmma_f32(ai, ai, pR);   // + Mi^T Mi
      pI = wmma_f32(ar, ai, pI);   // Mr^T Mi
      pI = wmma_f32(nai, ar, pI);  // - Mi^T Mr
    }
    const float t = (1.0f - exp2f(Rs)) / SNR;
#pragma unroll 8
    for (int g = 0; g < 8; ++g) {
      const int row = g + 8 * h;
      s.A1r[row * NT + m] = pR[g] + (row == m ? t : 0.0f);
      s.A1i[row * NT + m] = pI[g];
    }
  }

  // ---------------- A2 = alpha^2 H^T H + beta^2 I  (WMMA, K=32) ------------
  {
    v8f p2 = {};
#pragma unroll
    for (int kk = 0; kk < 8; ++kk) {
      const int k0 = kk * 4 + 2 * h;
      const int k1 = k0 + 1;
      v2f a;
      a.x = s.H[k0 * NT + m];
      a.y = s.H[k1 * NT + m];
      p2 = wmma_f32(a, a, p2);
    }
#pragma unroll 8
    for (int g = 0; g < 8; ++g) {
      const int row = g + 8 * h;
      s.L[row * NT + m] = a2 * p2[g] + (row == m ? b2 : 0.0f);
    }
  }
  __syncthreads();

  // ---------------- Cholesky of A2 (in place in s.L, lower) ----------------
  for (int j = 0; j < NT; ++j) {
    if (lane == j) {
      float d = s.L[j * NT + j];
      for (int k = 0; k < j; ++k) d -= s.L[j * NT + k] * s.L[j * NT + k];
      s.L[j * NT + j] = sqrtf(fmaxf(d, 1e-30f));
    }
    __syncthreads();
    if (lane > j && lane < NT) {
      float x = s.L[lane * NT + j];
      for (int k = 0; k < j; ++k) x -= s.L[lane * NT + k] * s.L[j * NT + k];
      s.L[lane * NT + j] = x / s.L[j * NT + j];
    }
    __syncthreads();
  }

  // ---------------- Z = L^{-1} A1 (forward solve; lane = column, h = part) -
  {
    float* Zp = h ? s.Zi : s.Zr;
    const float* Ap = h ? s.A1i : s.A1r;
    for (int i = 0; i < NT; ++i) {
      float x = Ap[i * NT + m];
      for (int k = 0; k < i; ++k) x -= s.L[i * NT + k] * Zp[k * NT + m];
      Zp[i * NT + m] = x / s.L[i * NT + i];
    }
  }
  __syncthreads();

  // ---------------- C = (L^{-1} Z^H)^H  (store C directly, transposed) -----
  {
    const float sgn = h ? -1.0f : 1.0f;
    const float* Zp = h ? s.Zi : s.Zr;
    float* Cp = h ? s.Ci : s.Cr;
    for (int i = 0; i < NT; ++i) {
      float x = sgn * Zp[m * NT + i];  // (Z^H)[i][m], this part
      for (int k = 0; k < i; ++k) x -= s.L[i * NT + k] * (sgn * Cp[m * NT + k]);
      x /= s.L[i * NT + i];
      Cp[m * NT + i] = sgn * x;  // Cr[m][i]=Wr[i][m], Ci[m][i]=-Wi[i][m]
    }
  }
  __syncthreads();

  // ---------------- power iteration: dominant |lambda| eigenvector of C ----
  if (h == 0) {
    s.vr[m] = 1.0f + 0.0625f * (float)m;
    s.vi[m] = 0.0f;
  }
  __syncthreads();
  for (int it = 0; it < 64; ++it) {
    float yr = 0.0f, yi = 0.0f;
    if (h == 0) {
      for (int k = 0; k < NT; ++k) {
        const float cr = s.Cr[m * NT + k], ci = s.Ci[m * NT + k];
        const float wr = s.vr[k], wi = s.vi[k];
        yr += cr * wr - ci * wi;
        yi += cr * wi + ci * wr;
      }
    }
    __syncthreads();
    s.red[lane] = (h == 0) ? (yr * yr + yi * yi) : 0.0f;
    __syncthreads();
    float nrm = 0.0f;
    for (int k = 0; k < NT; ++k) nrm += s.red[k];
    const float inv = rsqrtf(nrm + 1e-30f);
    if (h == 0) {
      s.vr[m] = yr * inv;
      s.vi[m] = yi * inv;
    }
    __syncthreads();
  }

  // ---------------- Beam = L^{-H} v  (backward solve) ----------------------
  {
    float sr = 0.0f, si = 0.0f;
    for (int i = NT - 1; i >= 0; --i) {
      if (lane == i) {
        const float d = s.L[i * NT + i];
        s.br[i] = (s.vr[i] - sr) / d;
        s.bi[i] = (s.vi[i] - si) / d;
      }
      __syncthreads();
      if (h == 0 && m < i) {
        const float lij = s.L[i * NT + m];
        sr += lij * s.br[i];
        si += lij * s.bi[i];
      }
      __syncthreads();
    }
  }
  // normalize Beam
  s.red[lane] = (h == 0) ? (s.br[m] * s.br[m] + s.bi[m] * s.bi[m]) : 0.0f;
  __syncthreads();
  {
    float nb = 0.0f;
    for (int k = 0; k < NT; ++k) nb += s.red[k];
    const float invb = rsqrtf(nb + 1e-30f);
    if (h == 0) {
      s.br[m] *= invb;
      s.bi[m] *= invb;
    }
  }
  __syncthreads();

  // ---------------- MB = M * Beam ; Cm -------------------------------------
  float myMB = 0.0f;
  if (lane < NR) {
    float mbr = 0.0f, mbi = 0.0f;
    for (int t_ = 0; t_ < NT; ++t_) {
      const float mr = s.Mr[lane * NT + t_], mi = s.Mi[lane * NT + t_];
      const float xr = s.br[t_], xi = s.bi[t_];
      mbr += mr * xr - mi * xi;
      mbi += mr * xi + mi * xr;
    }
    myMB = mbr * mbr + mbi * mbi;
  }
  s.red[lane] = myMB;
  __syncthreads();
  float sMB = 0.0f;
  for (int k = 0; k < NR; ++k) sMB += s.red[k];
  __syncthreads();

  // ---------------- |PhiH * Beam|^2 : |theta_s|=1 drops theta --------------
  {
    float hr = 0.0f, hi = 0.0f;
    for (int t_ = 0; t_ < NT; ++t_) {
      const float hv = s.H[lane * NT + t_];
      hr += hv * s.br[t_];
      hi += hv * s.bi[t_];
    }
    s.red[lane] = hr * hr + hi * hi;
  }
  __syncthreads();
  float sPhi = 0.0f;
  for (int k = 0; k < 32; ++k) sPhi += s.red[k];

  // ---------------- scalar tail + per-block partial ------------------------
  const float Cm = log2f(1.0f + SNR * sMB);
  const float phi1 = (exp2f(Cm - Rs) - 1.0f) / SNR;
  const float z0 = phi1 / (b2 + a2 * sPhi);

  if (lane == 0) blockAcc[wave] = -z0 * (1.0f / (float)BATCH);
  __syncthreads();
  if (tid == 0) {
    float tsum = 0.0f;
    for (int w = 0; w < WAVES; ++w) tsum += blockAcc[w];
    partial[blockIdx.x] = tsum;  // no atomics: deterministic 2-stage reduce
  }
}

__global__ __launch_bounds__(256) void reduce_kernel(
    const float* __restrict__ part, float* __restrict__ out) {
  __shared__ float acc[256];
  float v = 0.0f;
  for (int i = threadIdx.x; i < NBLOCKS; i += 256) v += part[i];
  acc[threadIdx.x] = v;
  __syncthreads();
  for (int s2 = 128; s2 > 0; s2 >>= 1) {
    if ((int)threadIdx.x < s2) acc[threadIdx.x] += acc[threadIdx.x + s2];
    __syncthreads();
  }
  if (threadIdx.x == 0) out[0] = acc[0];
}

extern "C" void kernel_launch(void* const* d_in, const int* in_sizes, int n_in,
                              void* d_out, int out_size, void* d_ws,
                              size_t ws_size, hipStream_t stream) {
  (void)in_sizes;
  (void)n_in;
  (void)out_size;
  (void)ws_size;
  const float* phase = (const float*)d_in[0];
  const float* target = (const float*)d_in[1];
  float* out = (float*)d_out;
  float* partial = (float*)d_ws;  // NBLOCKS floats = 32 KB scratch
  beam_loss_kernel<<<NBLOCKS, 32 * WAVES, 0, stream>>>(phase, target, partial);
  reduce_kernel<<<1, 256, 0, stream>>>(partial, out);
}